// ExplicitMC_60078002536492
// MI455X (gfx1250) — compile-verified
//
#include <hip/hip_runtime.h>
#include <stdint.h>
#include <math.h>

// ---------------- problem constants (match reference) ----------------
#define N_REACHES 131072
#define S_LEVEL   4096
#define L_LEVELS  32
#define T_STEPS   365
#define MAX_T     364          // time steps 1..364 computed by the recurrence
#define N_GAGES   16
#define BLOCK     256
#define WG_PER_LEVEL (S_LEVEL / BLOCK)   // 16
#define NWG (L_LEVELS * WG_PER_LEVEL)    // 512 workgroups, one reach per thread

typedef unsigned int u32;
typedef __attribute__((ext_vector_type(4))) u32 v4u;
typedef __attribute__((ext_vector_type(8))) int v8i;
typedef __attribute__((ext_vector_type(4))) int v4i;

#if __has_builtin(__builtin_amdgcn_tensor_load_to_lds)
#define HAVE_TDM 1
#else
#define HAVE_TDM 0
#endif

// ---------------- TDM: async 256-float row tile  global -> LDS ----------------
// D# descriptor per CDNA5 ISA:
//   group0: count=1 | lds_addr | global_addr[56:0] | type=2
//   group1: data_size=2 (4B), tensor_dim0=256, tensor_dim1=1,
//           tile_dim0=256, tile_dim1=1, tensor_dim0_stride=256
__device__ __forceinline__ void tdm_load_256f(const float* gsrc, u32 lds_byte_off) {
#if HAVE_TDM
  uint64_t ga = (uint64_t)(uintptr_t)gsrc;
  v4u g0 = { 1u,                                           // count=1 (valid D#)
             lds_byte_off,                                 // LDS dest (bytes)
             (u32)ga,                                      // global_addr[31:0]
             ((u32)(ga >> 32) & 0x01FFFFFFu) | 0x80000000u // addr[56:32] | type=2
           };
  v8i g1 = { (int)0x00020000u,      // wg_mask=0, data_size=2 (4 bytes)
             (int)(256u << 16),     // tensor_dim0[15:0] = 256   (bits 63:48)
             (int)(1u   << 16),     // tensor_dim0 hi=0, tensor_dim1[15:0]=1
             (int)(256u << 16),     // tensor_dim1 hi=0, tile_dim0=256 (bits 127:112)
             1,                     // tile_dim1=1, tile_dim2=0
             256,                   // tensor_dim0_stride[31:0]
             0, 0 };
  v4i z4 = {0, 0, 0, 0};
#if defined(__clang_major__) && (__clang_major__ >= 23)
  v8i z8 = {0, 0, 0, 0, 0, 0, 0, 0};
  __builtin_amdgcn_tensor_load_to_lds(g0, g1, z4, z4, z8, 0);   // therock 6-arg form
#else
  __builtin_amdgcn_tensor_load_to_lds(g0, g1, z4, z4, 0);       // ROCm 7.2 5-arg form
#endif
#else
  (void)gsrc; (void)lds_byte_off;
#endif
}

__device__ __forceinline__ void tdm_wait() {
#if __has_builtin(__builtin_amdgcn_s_wait_tensorcnt)
  __builtin_amdgcn_s_wait_tensorcnt(0);
#endif
}

// ---------------- device-wide sense-reversing barrier ----------------
// bar[0] = arrival counter, bar[16] = generation (separate cachelines).
__device__ __forceinline__ void grid_sync(u32* bar) {
  __syncthreads();
  if (threadIdx.x == 0) {
    __threadfence();   // release: make this WG's D writes device-visible
    u32* gen = bar + 16;
    u32 g = __hip_atomic_load(gen, __ATOMIC_RELAXED, __HIP_MEMORY_SCOPE_AGENT);
    u32 v = __hip_atomic_fetch_add(bar, 1u, __ATOMIC_ACQ_REL, __HIP_MEMORY_SCOPE_AGENT);
    if (v == (u32)(NWG - 1)) {
      __hip_atomic_store(bar, 0u, __ATOMIC_RELAXED, __HIP_MEMORY_SCOPE_AGENT);
      __hip_atomic_fetch_add(gen, 1u, __ATOMIC_RELEASE, __HIP_MEMORY_SCOPE_AGENT);
    } else {
      while (__hip_atomic_load(gen, __ATOMIC_ACQUIRE, __HIP_MEMORY_SCOPE_AGENT) == g) {
        __builtin_amdgcn_s_sleep(8);
      }
    }
    __threadfence();   // acquire: other WGs' D writes visible before we proceed
  }
  __syncthreads();
}

// ---------------- wavefront Muskingum-Cunge routing ----------------
// Cell (t, L) depends on (t, L-1), (t-1, L-1), (t-1, L): a 2D wavefront.
// Diagonal d processes all cells (t = d - L, level L) in parallel -> 397
// device-wide steps instead of 365*32 = 11,680 level-serial steps.
// Discharge state D is triple-buffered by t%3 (a buffer written at diagonal
// t+L is last read at t+L+2 and overwritten at t+L+3).
__global__ __launch_bounds__(BLOCK)
void muskingum_wavefront(const float* __restrict__ q_prime,   // [365][131072]
                         const float* __restrict__ attributes,// [131072][2]
                         const float* __restrict__ n_mannings,
                         const float* __restrict__ q_spatial,
                         const int*   __restrict__ upstream,  // [131072][2]
                         const int*   __restrict__ gages,     // [16][4]
                         float*       __restrict__ out,       // [16][365]
                         float*       __restrict__ D,         // ws: [3][131072]
                         u32*         __restrict__ bar)
{
  __shared__ float qbuf[2][BLOCK];    // double-buffered q_prime tile (TDM target)

  const int tid  = threadIdx.x;
  const int wg   = blockIdx.x;
  const int L    = wg / WG_PER_LEVEL;                       // topological level
  const int base = L * S_LEVEL + (wg % WG_PER_LEVEL) * BLOCK;
  const int r    = base + tid;                              // this thread's reach

  // ---- per-reach invariants, computed once, held in registers ----
  const float len = attributes[2 * r + 0];
  const float s0  = fminf(fmaxf(attributes[2 * r + 1], 1e-4f), 0.3f);
  const float n   = n_mannings[r];
  const float q   = q_spatial[r];
  const float sq  = __builtin_sqrtf(s0);
  const float A   = n * (q + 1.0f) / (21.0f * sq);   // depth arg scale (P_SPATIAL=21)
  const float e23 = 2.0f / (5.0f + 3.0f * q);        // v = B * (q_t*A)^e23
  const float Bv  = sq / n;
  const int   u0  = upstream[2 * r + 0];
  const int   u1  = upstream[2 * r + 1];

  int g0i = 0, g1i = 0, g2i = 0, g3i = 0;
  if (wg == 0 && tid < N_GAGES) {
    g0i = gages[tid * 4 + 0]; g1i = gages[tid * 4 + 1];
    g2i = gages[tid * 4 + 2]; g3i = gages[tid * 4 + 3];
  }

  // ---- t = 0: discharge0 = q_prime[0]; emit gage column 0 ----
  D[r] = q_prime[r];                       // buffer (t=0)%3 == 0
  if (wg == 0 && tid < N_GAGES) {
    out[tid * T_STEPS + 0] =
        q_prime[g0i] + q_prime[g1i] + q_prime[g2i] + q_prime[g3i];
  }

#if HAVE_TDM
  // Prefetch q_prime[1] tile; it is consumed at this WG's first active
  // diagonal d = L+1 (TDM is a wave-level op: issue from wave 0 only).
  if (tid == 0)
    tdm_load_256f(q_prime + 1 * N_REACHES + base, (u32)(uintptr_t)&qbuf[1][0]);
#endif

  grid_sync(bar);

  // Compute runs d = 1..395; gage readout trails level 31 by one diagonal.
  for (int d = 1; d <= MAX_T + L_LEVELS; ++d) {
    const int t = d - L;
    const bool active = (t >= 1 && t <= MAX_T);

#if HAVE_TDM
    if (active && tid == 0) {
      tdm_wait();                                   // qbuf[t&1] is resident
      if (t < MAX_T)                                // prefetch next tile (hidden
        tdm_load_256f(q_prime + (size_t)(t + 1) * N_REACHES + base,   // behind
                      (u32)(uintptr_t)&qbuf[(t + 1) & 1][0]);         // compute+sync)
    }
#endif
    __syncthreads();

    if (active) {
      const float* Dp = D + (u32)((t - 1) % 3) * N_REACHES;  // discharge_t
      float*       Dc = D + (u32)(t % 3)       * N_REACHES;  // discharge_{t+1}
      const float  q_t = Dp[r];
#if HAVE_TDM
      const float qp = qbuf[t & 1][tid];
#else
      const float qp = q_prime[(size_t)t * N_REACHES + r];
#endif
      float it = 0.0f, it1 = 0.0f;
      if (u0 >= 0) { it += Dp[u0]; it1 += Dc[u0]; }  // level L-1: written at d-2 / d-1
      if (u1 >= 0) { it += Dp[u1]; it1 += Dc[u1]; }

      // v = (1/n) * depth^(2/3) * sqrt(s0), depth = (q_t*A)^(3/(5+3q))
      //   = Bv * (q_t*A)^(2/(5+3q));  libm powf for accuracy over 364 steps.
      const float v   = Bv * powf(q_t * A, e23);
      const float c   = fminf(fmaxf(v, 0.3f), 15.0f) * (5.0f / 3.0f);
      const float k   = len / c;
      const float den = 1.4f * k + 3600.0f;          // 2k(1-X)+T, X=0.3, T=3600
      const float rdn = 1.0f / den;
      const float c1  = (3600.0f - 0.6f * k) * rdn;
      const float c2  = (3600.0f + 0.6f * k) * rdn;
      const float c3  = (1.4f * k - 3600.0f) * rdn;
      const float c4  = 7200.0f * rdn;
      Dc[r] = fmaxf(c1 * it1 + c2 * it + c3 * q_t + c4 * qp, 1e-4f);
    }

    // Gage readout for the timestep whose level-31 cells finished at d-1.
    if (wg == 0) {
      const int to = d - L_LEVELS;
      if (to >= 1 && to <= MAX_T && tid < N_GAGES) {
        const float* Db = D + (u32)(to % 3) * N_REACHES;
        out[tid * T_STEPS + to] = Db[g0i] + Db[g1i] + Db[g2i] + Db[g3i];
      }
    }

    grid_sync(bar);
  }
}

// ---------------- host entry ----------------
extern "C" void kernel_launch(void* const* d_in, const int* in_sizes, int n_in,
                              void* d_out, int out_size, void* d_ws, size_t ws_size,
                              hipStream_t stream) {
  (void)in_sizes; (void)n_in; (void)out_size; (void)ws_size;
  const float* q_prime = (const float*)d_in[0];
  const float* attrs   = (const float*)d_in[1];
  const float* n_man   = (const float*)d_in[2];
  const float* q_spat  = (const float*)d_in[3];
  // d_in[4] index_graph is just arange(N).reshape(32,4096): level derived from id.
  const int*   upstrm  = (const int*)d_in[5];
  const int*   gages   = (const int*)d_in[6];
  float* out = (float*)d_out;

  float* D  = (float*)d_ws;                                    // 3*131072 f32
  u32*  bar = (u32*)((char*)d_ws + (size_t)3 * N_REACHES * sizeof(float));

  // Barrier words must start at 0 every call (ws is poisoned once, never
  // restored). Async memset is graph-capture legal (becomes a memset node).
  hipMemsetAsync(bar, 0, 128, stream);

  hipLaunchKernelGGL(muskingum_wavefront, dim3(NWG), dim3(BLOCK), 0, stream,
                     q_prime, attrs, n_man, q_spat, upstrm, gages, out, D, bar);
}